// MetaGCN_69157563400217
// MI455X (gfx1250) — compile-verified
//
#include <hip/hip_runtime.h>

typedef __attribute__((ext_vector_type(2))) float v2f;
typedef __attribute__((ext_vector_type(8))) float v8f;

#define FEAT 64
#define BN_EPS 1e-5f

__device__ __forceinline__ void atomAddF(float* p, float v) {
    __hip_atomic_fetch_add(p, v, __ATOMIC_RELAXED, __HIP_MEMORY_SCOPE_AGENT);
}

// ---------------- degree / normalization ----------------

__global__ void k_deg_init(float* __restrict__ deg, int N) {
    int i = blockIdx.x * blockDim.x + threadIdx.x;
    if (i < N) deg[i] = 1.0f;   // appended self-loop contributes weight 1
}

__global__ void k_deg_scatter(const int* __restrict__ ei, int E, float* __restrict__ deg) {
    int e = blockIdx.x * blockDim.x + threadIdx.x;
    if (e >= E) return;
    int r = ei[e];
    int c = ei[E + e];
    if (r != c) atomAddF(&deg[c], 1.0f);   // original self-loops have weight 0
}

__global__ void k_dinv(float* __restrict__ deg, float* __restrict__ stats, int N) {
    int i = blockIdx.x * blockDim.x + threadIdx.x;
    if (i < 128) stats[i] = 0.0f;          // zero BN sum/sumsq scratch
    if (i < N) {
        float d = deg[i];
        deg[i] = (d > 0.0f) ? rsqrtf(d) : 0.0f;   // in place: deg -> d^-1/2
    }
}

// ---------------- GEMM: h = x @ W^T  via V_WMMA_F32_16X16X4_F32 ----------------
// Each wave: 16 rows of x, all 64 output cols (4 accumulators of 16x16).
// A(16x4 f32) layout: lanes 0-15 hold M=0..15; VGPR0/1 = K pair (k0+2*hi, +1).
// B(4x16 f32) layout (mirrored): lanes hold N=0..15; VGPR0/1 = K pair per half.
// C/D(16x16 f32): VGPR r = row (r + 8*hi), lane&15 = col.

__global__ __launch_bounds__(256) void k_gemm(const float* __restrict__ x,
                                              const float* __restrict__ W,
                                              float* __restrict__ h, int N) {
    __shared__ float Ws[64 * 66];          // pad 64->66: fragment reads hit distinct banks
    for (int t = threadIdx.x; t < 64 * 64; t += 256) {
        int j = t >> 6, k = t & 63;
        Ws[j * 66 + k] = W[t];             // Ws[j][k] = W[j,k]  (B[k][j] = W[j][k])
    }
    __syncthreads();

    const int wave = threadIdx.x >> 5;
    const int lane = threadIdx.x & 31;
    const int m    = lane & 15;            // row within A tile / col within B,C tiles
    const int hi   = lane >> 4;            // half-wave select
    const int r0   = blockIdx.x * 128 + wave * 16;

    int rowA  = r0 + m;
    int rowAc = rowA < N ? rowA : (N - 1); // clamp: keeps EXEC all-ones for WMMA
    const float* xrow = x + (size_t)rowAc * FEAT;

    v8f c0 = {}, c1 = {}, c2 = {}, c3 = {};
    #pragma unroll
    for (int k0 = 0; k0 < 64; k0 += 4) {
        int kb = k0 + hi * 2;
        v2f a  = *(const v2f*)(xrow + kb);
        v2f b0 = *(const v2f*)&Ws[(0  + m) * 66 + kb];
        v2f b1 = *(const v2f*)&Ws[(16 + m) * 66 + kb];
        v2f b2 = *(const v2f*)&Ws[(32 + m) * 66 + kb];
        v2f b3 = *(const v2f*)&Ws[(48 + m) * 66 + kb];
        c0 = __builtin_amdgcn_wmma_f32_16x16x4_f32(false, a, false, b0, (short)0, c0, false, false);
        c1 = __builtin_amdgcn_wmma_f32_16x16x4_f32(false, a, false, b1, (short)0, c1, false, false);
        c2 = __builtin_amdgcn_wmma_f32_16x16x4_f32(false, a, false, b2, (short)0, c2, false, false);
        c3 = __builtin_amdgcn_wmma_f32_16x16x4_f32(false, a, false, b3, (short)0, c3, false, false);
    }

    #pragma unroll
    for (int r = 0; r < 8; ++r) {
        int row = r0 + r + hi * 8;
        if (row < N) {
            float* hr = h + (size_t)row * FEAT + m;
            hr[0]  = c0[r];
            hr[16] = c1[r];
            hr[32] = c2[r];
            hr[48] = c3[r];
        }
    }
}

// ---------------- out = bias + h * dinv^2 (self-loop term) ----------------

__global__ void k_init_out(const float* __restrict__ h, const float* __restrict__ dinv,
                           const float* __restrict__ bias, float* __restrict__ out,
                           long long total) {
    long long i = blockIdx.x * (long long)blockDim.x + threadIdx.x;
    if (i >= total) return;
    int n = (int)(i >> 6);
    int d = (int)(i & 63);
    float w = dinv[n];
    out[i] = bias[d] + h[i] * w * w;
}

// ---------------- edge scatter: 32 lanes per edge ----------------

__global__ __launch_bounds__(256) void k_scatter(const int* __restrict__ ei, long long E,
                                                 const float* __restrict__ h,
                                                 const float* __restrict__ dinv,
                                                 float* __restrict__ out) {
    long long gt = blockIdx.x * (long long)blockDim.x + threadIdx.x;
    long long e  = gt >> 5;
    if (e >= E) return;
    int lane = threadIdx.x & 31;
    int r = ei[e];
    int c = ei[E + e];
    if (r == c) return;                    // original self-loops carry weight 0
    float w = dinv[r] * dinv[c];
    const float* hr = h + (size_t)r * FEAT;
    float* oc = out + (size_t)c * FEAT;
    atomAddF(&oc[lane],      hr[lane]      * w);
    atomAddF(&oc[lane + 32], hr[lane + 32] * w);
}

// ---------------- BN statistics: per-column sum / sumsq ----------------

__global__ __launch_bounds__(256) void k_stats(const float* __restrict__ out,
                                               float* __restrict__ stats, long long total) {
    __shared__ float s1[256];
    __shared__ float s2[256];
    int t = threadIdx.x;                   // column = t & 63 (stride multiple of 64)
    float a1 = 0.0f, a2 = 0.0f;
    for (long long i = blockIdx.x * 256LL + t; i < total; i += (long long)gridDim.x * 256LL) {
        float v = out[i];
        a1 += v;
        a2 += v * v;
    }
    s1[t] = a1; s2[t] = a2;
    __syncthreads();
    if (t < 128) { s1[t] += s1[t + 128]; s2[t] += s2[t + 128]; }
    __syncthreads();
    if (t < 64) {
        atomAddF(&stats[t],      s1[t] + s1[t + 64]);
        atomAddF(&stats[64 + t], s2[t] + s2[t + 64]);
    }
}

// ---------------- BN apply + PReLU (in place on d_out) ----------------

__global__ void k_bn_prelu(float* __restrict__ out, const float* __restrict__ stats,
                           const float* __restrict__ bnw, const float* __restrict__ bnb,
                           const float* __restrict__ prelu, float invN, long long total) {
    long long i = blockIdx.x * (long long)blockDim.x + threadIdx.x;
    if (i >= total) return;
    int d = (int)(i & 63);
    float mean = stats[d] * invN;
    float var  = stats[64 + d] * invN - mean * mean;   // biased variance
    float v = (out[i] - mean) * rsqrtf(var + BN_EPS) * bnw[d] + bnb[d];
    out[i] = (v >= 0.0f) ? v : prelu[0] * v;
}

// ---------------- host launcher ----------------

extern "C" void kernel_launch(void* const* d_in, const int* in_sizes, int n_in,
                              void* d_out, int out_size, void* d_ws, size_t ws_size,
                              hipStream_t stream) {
    const float* x     = (const float*)d_in[0];
    const int*   ei    = (const int*)d_in[1];
    const float* W     = (const float*)d_in[2];
    const float* bias  = (const float*)d_in[3];
    const float* bnw   = (const float*)d_in[4];
    const float* bnb   = (const float*)d_in[5];
    const float* prelu = (const float*)d_in[6];

    const int N       = in_sizes[0] / FEAT;
    const long long E = in_sizes[1] / 2;
    const long long total = (long long)N * FEAT;

    float* dinv  = (float*)d_ws;       // N floats
    float* stats = dinv + N;           // 128 floats (sum | sumsq)
    float* h     = stats + 128;        // N*64 floats
    float* out   = (float*)d_out;

    k_deg_init   <<<(N + 255) / 256, 256, 0, stream>>>(dinv, N);
    k_deg_scatter<<<(int)((E + 255) / 256), 256, 0, stream>>>(ei, (int)E, dinv);
    k_dinv       <<<(N + 255) / 256, 256, 0, stream>>>(dinv, stats, N);
    k_gemm       <<<(N + 127) / 128, 256, 0, stream>>>(x, W, h, N);
    k_init_out   <<<(int)((total + 255) / 256), 256, 0, stream>>>(h, dinv, bias, out, total);
    long long sthreads = E * 32;
    k_scatter    <<<(int)((sthreads + 255) / 256), 256, 0, stream>>>(ei, E, h, dinv, out);
    k_stats      <<<1024, 256, 0, stream>>>(out, stats, total);
    k_bn_prelu   <<<(int)((total + 255) / 256), 256, 0, stream>>>(out, stats, bnw, bnb, prelu,
                                                                  1.0f / (float)N, total);
}